// OrgModel_4999341932625
// MI455X (gfx1250) — compile-verified
//
#include <hip/hip_runtime.h>

#define USERN 50000
#define NN    100000
#define DD    64
#define HH    128
#define EE    1600000
#define SLOPE 0.5f
#define KCHUNK 1024

typedef float v2f __attribute__((ext_vector_type(2)));
typedef float v8f __attribute__((ext_vector_type(8)));

static __device__ __forceinline__ v8f wmma4(v2f a, v2f b, v8f c) {
  // V_WMMA_F32_16X16X4_F32 : D = A(16x4) x B(4x16) + C  (fp32 exact)
  return __builtin_amdgcn_wmma_f32_16x16x4_f32(false, a, false, b, (short)0, c,
                                               false, false);
}
static __device__ __forceinline__ float leaky(float x) {
  return x > 0.f ? x : SLOPE * x;
}

// ---------------- elementwise kernels ----------------

__global__ __launch_bounds__(256) void k_zero(float* __restrict__ p, int n4) {
  int i = blockIdx.x * blockDim.x + threadIdx.x;
  if (i < n4) ((float4*)p)[i] = make_float4(0.f, 0.f, 0.f, 0.f);
}

__global__ __launch_bounds__(256) void k_concat(const float* __restrict__ u,
                                                const float* __restrict__ it,
                                                float* __restrict__ cur) {
  int i = blockIdx.x * blockDim.x + threadIdx.x;   // float4 index
  const int uN4 = USERN * DD / 4;
  const int tot4 = NN * DD / 4;
  if (i >= tot4) return;
  float4 v = (i < uN4) ? ((const float4*)u)[i] : ((const float4*)it)[i - uN4];
  ((float4*)cur)[i] = v;
}

// xg = cur * (2*sigmoid(z) - 1)
__global__ __launch_bounds__(256) void k_gate(const float* __restrict__ cur,
                                              const float* __restrict__ z,
                                              float* __restrict__ xg, int n) {
  int i = blockIdx.x * blockDim.x + threadIdx.x;
  if (i < n) {
    float s = 1.0f / (1.0f + __expf(-z[i]));
    xg[i] = cur[i] * (2.0f * s - 1.0f);
  }
}

// tem = leaky(accum); cur = hyperLat + tem   (temOut may alias accum)
__global__ __launch_bounds__(256) void k_epilogue(const float* __restrict__ accum,
                                                  const float* __restrict__ hyperLat,
                                                  float* __restrict__ temOut,
                                                  float* __restrict__ curOut, int n) {
  int i = blockIdx.x * blockDim.x + threadIdx.x;
  if (i < n) {
    float t = leaky(accum[i]);
    float h = hyperLat[i];
    temOut[i] = t;
    curOut[i] = h + t;
  }
}

// ---------------- SpMM: accum[r,:] += val * xg[c,:] ----------------
// one wave per edge; 32 lanes cover D=64 as float2. Gathers are coalesced,
// scatter via global_atomic_add_f32 (L2-resident).
__global__ __launch_bounds__(256) void k_spmm(const int* __restrict__ rows,
                                              const int* __restrict__ cols,
                                              const float* __restrict__ vals,
                                              const float* __restrict__ xg,
                                              float* __restrict__ accum, int nE) {
  int lane = threadIdx.x & 31;
  int wid = (blockIdx.x * blockDim.x + threadIdx.x) >> 5;
  int nW = (gridDim.x * blockDim.x) >> 5;
  for (int e = wid; e < nE; e += nW) {
    int r = rows[e];
    int c = cols[e];
    float v = vals[e];
    float2 x = *(const float2*)(xg + (size_t)c * DD + 2 * lane);
    float* dst = accum + (size_t)r * DD + 2 * lane;
    atomicAdd(dst + 0, v * x.x);
    atomicAdd(dst + 1, v * x.y);
  }
}

// ---------------- GEMM1: allHyper[N,H] = cur[N,64] @ Hyper[64,128] ----------------
// block = 8 waves, each wave owns one 16-wide H tile; blockIdx = M tile (N/16).
__global__ __launch_bounds__(256) void k_gemm_embeds_hyper(
    const float* __restrict__ cur, const float* __restrict__ Hyp,
    float* __restrict__ out) {
  int wave = threadIdx.x >> 5;          // 0..7 -> H tile
  int lane = threadIdx.x & 31;
  int half = lane >> 4, l16 = lane & 15;
  int mt = blockIdx.x;                  // 0..N/16-1
  const float* A0 = cur + (size_t)(mt * 16 + l16) * DD + 2 * half;
  const float* B0 = Hyp + (size_t)(2 * half) * HH + wave * 16 + l16;
  v8f c = {};
#pragma unroll
  for (int k = 0; k < DD; k += 4) {
    v2f a; a.x = A0[k]; a.y = A0[k + 1];
    v2f b; b.x = B0[k * HH]; b.y = B0[(k + 1) * HH];
    c = wmma4(a, b, c);
  }
  float* Dp = out + (size_t)(mt * 16) * HH + wave * 16 + l16;
#pragma unroll
  for (int v = 0; v < 8; ++v) Dp[(size_t)(v + 8 * half) * HH] = c[v];
}

// ---------------- GEMM2: latRaw[128,64] += hyp.T @ cur  (split-K over N) ------
// blockIdx = K chunk; 8 waves each own one H tile x all 4 D tiles; partials
// merged with f32 atomics into the tiny latRaw buffer.
__global__ __launch_bounds__(256) void k_gemm_hypT_cur(
    const float* __restrict__ hyp, const float* __restrict__ cur,
    float* __restrict__ latRaw) {
  int wave = threadIdx.x >> 5;          // H tile 0..7
  int lane = threadIdx.x & 31;
  int half = lane >> 4, l16 = lane & 15;
  int k0 = blockIdx.x * KCHUNK;
  int kend = k0 + KCHUNK; if (kend > NN) kend = NN;   // tail is multiple of 4
  v8f c0 = {}, c1 = {}, c2 = {}, c3 = {};
  for (int k = k0; k < kend; k += 4) {
    int kg = k + 2 * half;
    // A[m][k] = hyp[k][m], m = wave*16 + l16
    v2f a;
    a.x = hyp[(size_t)kg * HH + wave * 16 + l16];
    a.y = hyp[(size_t)(kg + 1) * HH + wave * 16 + l16];
    const float* r0 = cur + (size_t)kg * DD;
    const float* r1 = cur + (size_t)(kg + 1) * DD;
    v2f b;
    b.x = r0[0 * 16 + l16]; b.y = r1[0 * 16 + l16]; c0 = wmma4(a, b, c0);
    b.x = r0[1 * 16 + l16]; b.y = r1[1 * 16 + l16]; c1 = wmma4(a, b, c1);
    b.x = r0[2 * 16 + l16]; b.y = r1[2 * 16 + l16]; c2 = wmma4(a, b, c2);
    b.x = r0[3 * 16 + l16]; b.y = r1[3 * 16 + l16]; c3 = wmma4(a, b, c3);
  }
  v8f acc[4] = {c0, c1, c2, c3};
#pragma unroll
  for (int nd = 0; nd < 4; ++nd)
#pragma unroll
    for (int v = 0; v < 8; ++v)
      atomicAdd(&latRaw[(size_t)(wave * 16 + v + 8 * half) * DD + nd * 16 + l16],
                acc[nd][v]);
}

// ---------------- GEMM3: out[N,64] = leaky( hyp[N,128] @ leaky(latRaw) ) ------
// lat staged in LDS (32KB) with leaky fused on the fill; leaky fused on store.
__global__ __launch_bounds__(256) void k_gemm_hyp_lat(
    const float* __restrict__ hyp, const float* __restrict__ latRaw,
    float* __restrict__ outSlot) {
  __shared__ float latS[HH * DD];
  for (int i = threadIdx.x; i < HH * DD; i += blockDim.x)
    latS[i] = leaky(latRaw[i]);
  __syncthreads();
  int wave = threadIdx.x >> 5;
  int mi = wave >> 2, nd = wave & 3;    // 2 M tiles x 4 D tiles per block
  int lane = threadIdx.x & 31;
  int half = lane >> 4, l16 = lane & 15;
  size_t row = (size_t)blockIdx.x * 32 + mi * 16 + l16;
  const float* A0 = hyp + row * HH + 2 * half;
  v8f c = {};
#pragma unroll
  for (int k = 0; k < HH; k += 4) {
    v2f a; a.x = A0[k]; a.y = A0[k + 1];
    v2f b;
    b.x = latS[(k + 2 * half) * DD + nd * 16 + l16];
    b.y = latS[(k + 2 * half + 1) * DD + nd * 16 + l16];
    c = wmma4(a, b, c);
  }
  size_t orow = (size_t)blockIdx.x * 32 + mi * 16;
  float* Dp = outSlot + orow * DD + nd * 16 + l16;
#pragma unroll
  for (int v = 0; v < 8; ++v)
    Dp[(size_t)(v + 8 * half) * DD] = leaky(c[v]);
}

// ---------------- launch ----------------

extern "C" void kernel_launch(void* const* d_in, const int* in_sizes, int n_in,
                              void* d_out, int out_size, void* d_ws, size_t ws_size,
                              hipStream_t stream) {
  const int*   rows = (const int*)d_in[0];
  const int*   cols = (const int*)d_in[1];
  const float* vals = (const float*)d_in[2];
  const float* uEmb = (const float*)d_in[3];
  const float* iEmb = (const float*)d_in[4];
  const float* Hyp  = (const float*)d_in[5];
  const float* zish = (const float*)d_in[6];
  // d_in[7] = keepRate == 1 -> dropout branch statically dead

  float* out = (float*)d_out;                 // [3, N, 64]
  float* accum = out;                          // slot 0 doubles as SpMM accum

  float* ws = (float*)d_ws;
  float* cur      = ws;                        // [N,64]
  float* allHyper = cur + (size_t)NN * DD;     // [N,128]
  float* xg       = allHyper + (size_t)NN * HH;// [N,64] (also tem scratch L0)
  float* latRaw   = xg + (size_t)NN * DD;      // [128,64]

  const int ND   = NN * DD;                    // 6,400,000
  const int ND4  = ND / 4;                     // 1,600,000
  const int HD4  = HH * DD / 4;                // 2,048
  const int eltB = (ND + 255) / 256;           // elementwise grid
  const int v4B  = (ND4 + 255) / 256;

  k_concat<<<v4B, 256, 0, stream>>>(uEmb, iEmb, cur);
  k_gemm_embeds_hyper<<<NN / 16, 256, 0, stream>>>(cur, Hyp, allHyper);

  for (int l = 0; l < 2; ++l) {
    k_gate<<<eltB, 256, 0, stream>>>(cur, zish, xg, ND);
    k_zero<<<v4B, 256, 0, stream>>>(accum, ND4);
    k_zero<<<(HD4 + 255) / 256, 256, 0, stream>>>(latRaw, HD4);
    k_spmm<<<2048, 256, 0, stream>>>(rows, cols, vals, xg, accum, EE);
    k_gemm_hypT_cur<<<(NN + KCHUNK - 1) / KCHUNK, 256, 0, stream>>>(allHyper, cur,
                                                                    latRaw);
    float* hl = out + (size_t)(1 + l) * ND;
    k_gemm_hyp_lat<<<NN / 32, 256, 0, stream>>>(allHyper, latRaw, hl);
    float* temDst = (l == 1) ? accum : xg;     // final tem lands in slot 0
    k_epilogue<<<eltB, 256, 0, stream>>>(accum, hl, temDst, cur, ND);
  }
}